// KWinners2d_85796266705113
// MI455X (gfx1250) — compile-verified
//
#include <hip/hip_runtime.h>
#include <stdint.h>

// ---------------------------------------------------------------------------
// KWinners2d for MI455X (gfx1250), compile-only tuned.
// Radix-select (11+11+10 bits) on monotone float keys, per batch row.
// LDS histograms via ds_add_u32, b128 streaming loads, optional async
// global->LDS staging of the boost table, global_prefetch hints.
// ---------------------------------------------------------------------------

#define KW_B        32
#define KW_N        401408                 // 56*56*128
#define KW_C        128
#define KW_THREADS  256
#define KW_SPLIT    56                     // blocks per row for histogram pass
#define KW_CHUNK    (KW_N / KW_SPLIT)      // 7168 elements per block
#define KW_F4_ITER  (KW_CHUNK / 4 / KW_THREADS)   // 7 float4 per thread
#define KW_MASK_BLK (KW_N / (4 * KW_THREADS))     // 392 blocks per row
#define KW_NBINS_MAX 2048

// workspace layout, in 32-bit words
enum {
  WSW_BOOST  = 0,                       // 128 floats: per-channel boost factor
  WSW_PREFIX = 128,                     // 32: decided high bits of threshold key
  WSW_KREM   = 160,                     // 32: winners remaining among candidates
  WSW_TIE    = 192,                     // 32: tie-admission counters
  WSW_HIST   = 256,                     // 32 * 2048 histogram bins
  WSW_TOTAL  = WSW_HIST + KW_B * KW_NBINS_MAX
};

// ---- optional CDNA5 async global->LDS path (probe-safe via __has_builtin) --
#if defined(__has_builtin)
#  if __has_builtin(__builtin_amdgcn_global_load_async_to_lds_b128) && \
      __has_builtin(__builtin_amdgcn_s_wait_asynccnt)
#    define KW_HAVE_ASYNC 1
#  endif
#endif
#ifndef KW_HAVE_ASYNC
#  define KW_HAVE_ASYNC 0
#endif

typedef int kw_v4i __attribute__((ext_vector_type(4)));
typedef __attribute__((address_space(1))) kw_v4i* kw_gptr;   // global int4*
typedef __attribute__((address_space(3))) kw_v4i* kw_lptr;   // LDS int4*

// Monotone map: float -> uint32 such that key order == float order.
__device__ __forceinline__ uint32_t kw_key(float f) {
  uint32_t u = __float_as_uint(f);
  return (u & 0x80000000u) ? ~u : (u | 0x80000000u);
}

// Stage the 128-entry boost table into LDS (async when the toolchain has it).
__device__ __forceinline__ void kw_stage_boost(const float* __restrict__ gb,
                                               float* lb, int t) {
#if KW_HAVE_ASYNC
  if (t < KW_C / 4) {
    __builtin_amdgcn_global_load_async_to_lds_b128(
        (kw_gptr)(gb + t * 4),
        (kw_lptr)(lb + t * 4),
        0, 0);
  }
  __builtin_amdgcn_s_wait_asynccnt(0);
#else
  if (t < KW_C) lb[t] = gb[t];
#endif
  __syncthreads();
}

// ---------------------------------------------------------------------------
// Kernel 0: compute boost factors, reset select state, zero histograms.
// ---------------------------------------------------------------------------
__global__ void __launch_bounds__(KW_THREADS)
kw_init(const float* __restrict__ duty, const float* __restrict__ bs_p,
        const int* __restrict__ k_p, uint32_t* __restrict__ ws) {
  const int t = threadIdx.x;
  const float bs = fmaxf(bs_p[0], 0.0f);
  const int k = k_p[0];
  const float td = (float)k / (float)KW_N;
  if (t < KW_C) {
    ((float*)ws)[WSW_BOOST + t] = expf((td - duty[t]) * bs);
  }
  if (t < KW_B) {
    ws[WSW_PREFIX + t] = 0u;
    ws[WSW_KREM + t]   = (uint32_t)k;
    ws[WSW_TIE + t]    = 0u;
  }
  for (int i = t; i < KW_B * KW_NBINS_MAX; i += KW_THREADS)
    ws[WSW_HIST + i] = 0u;
}

// ---------------------------------------------------------------------------
// Histogram pass: per-row digit histogram of boosted keys among candidates
// matching the decided prefix. LDS histogram (ds_add_u32) -> global merge.
// ---------------------------------------------------------------------------
__global__ void __launch_bounds__(KW_THREADS)
kw_hist(const float* __restrict__ x, uint32_t* __restrict__ ws,
        int shift, uint32_t bin_mask, uint32_t hi_mask) {
  __shared__ uint32_t lhist[KW_NBINS_MAX];
  __shared__ float    lboost[KW_C];
  const int t   = threadIdx.x;
  const int row = blockIdx.y;

  for (int i = t; i < KW_NBINS_MAX; i += KW_THREADS) lhist[i] = 0u;
  kw_stage_boost((const float*)ws + WSW_BOOST, lboost, t);  // has barrier

  const uint32_t pref = ws[WSW_PREFIX + row];
  const float4* xr =
      (const float4*)(x + (size_t)row * KW_N + (size_t)blockIdx.x * KW_CHUNK);
  const int ebase = blockIdx.x * KW_CHUNK;

#pragma unroll
  for (int i = 0; i < KW_F4_ITER; ++i) {
    const int f4 = i * KW_THREADS + t;
    if (i + 1 < KW_F4_ITER)
      __builtin_prefetch(&xr[f4 + KW_THREADS], 0, 0);   // global_prefetch_b8
    const float4 v = xr[f4];
    const int c = (ebase + f4 * 4) & (KW_C - 1);        // channel of lane 0 of the quad
    uint32_t kk[4];
    kk[0] = kw_key(v.x * lboost[c + 0]);
    kk[1] = kw_key(v.y * lboost[c + 1]);
    kk[2] = kw_key(v.z * lboost[c + 2]);
    kk[3] = kw_key(v.w * lboost[c + 3]);
#pragma unroll
    for (int j = 0; j < 4; ++j)
      if ((kk[j] & hi_mask) == pref)
        atomicAdd(&lhist[(kk[j] >> shift) & bin_mask], 1u);   // ds_add_u32
  }
  __syncthreads();

  uint32_t* gh = ws + WSW_HIST + row * KW_NBINS_MAX;
  for (int i = t; i < KW_NBINS_MAX; i += KW_THREADS) {
    const uint32_t cnt = lhist[i];
    if (cnt) atomicAdd(&gh[i], cnt);                          // global_atomic_add
  }
}

// ---------------------------------------------------------------------------
// Select pass: scan bins high->low, pick the digit where the cumulative count
// crosses kremain; update prefix/kremain; zero histogram for the next pass.
// One block per row; partial sums parallel, crossing search serial in LDS.
// ---------------------------------------------------------------------------
__global__ void __launch_bounds__(KW_THREADS)
kw_select(uint32_t* __restrict__ ws, int shift, int nbins, int is_last) {
  __shared__ uint32_t lh[KW_NBINS_MAX];
  __shared__ uint32_t psum[KW_THREADS];
  const int t   = threadIdx.x;
  const int row = blockIdx.x;
  uint32_t* gh = ws + WSW_HIST + row * KW_NBINS_MAX;
  const int per = nbins / KW_THREADS;     // 8 (11-bit) or 4 (10-bit)

  uint32_t s = 0;
  for (int i = 0; i < per; ++i) {
    const uint32_t c = gh[t * per + i];
    lh[t * per + i] = c;
    s += c;
  }
  psum[t] = s;
  __syncthreads();

  if (t == 0) {
    const uint32_t kr = ws[WSW_KREM + row];
    uint32_t cum = 0;
    int seg = KW_THREADS - 1;
    for (; seg > 0; --seg) {
      if (cum + psum[seg] >= kr) break;
      cum += psum[seg];
    }
    int d = seg * per;
    for (int b = seg * per + per - 1; b >= seg * per; --b) {
      const uint32_t c = lh[b];
      if (cum + c >= kr) { d = b; break; }
      cum += c;
    }
    ws[WSW_PREFIX + row] |= ((uint32_t)d) << shift;
    ws[WSW_KREM + row] = kr - cum;      // winners inside chosen bin
    if (is_last) ws[WSW_TIE + row] = 0u;
  }
  __syncthreads();
  for (int i = t; i < KW_NBINS_MAX; i += KW_THREADS) gh[i] = 0u;
}

// ---------------------------------------------------------------------------
// Mask pass: out = x where key > T; admit exactly kremain elements with
// key == T (per-row atomic counter); else 0.
// ---------------------------------------------------------------------------
__global__ void __launch_bounds__(KW_THREADS)
kw_mask(const float* __restrict__ x, uint32_t* __restrict__ ws,
        float* __restrict__ out) {
  __shared__ float lboost[KW_C];
  const int t   = threadIdx.x;
  const int row = blockIdx.y;
  kw_stage_boost((const float*)ws + WSW_BOOST, lboost, t);

  const uint32_t T     = ws[WSW_PREFIX + row];
  const uint32_t nties = ws[WSW_KREM + row];
  uint32_t* tie = ws + WSW_TIE + row;

  const int e_in_row = (blockIdx.x * KW_THREADS + t) * 4;
  const size_t e = (size_t)row * KW_N + (size_t)e_in_row;
  const float4 v = *(const float4*)(x + e);
  const int c = e_in_row & (KW_C - 1);

  float in[4] = {v.x, v.y, v.z, v.w};
  float bf[4] = {lboost[c], lboost[c + 1], lboost[c + 2], lboost[c + 3]};
  float4 o;
  float* op = &o.x;
#pragma unroll
  for (int j = 0; j < 4; ++j) {
    const uint32_t kk = kw_key(in[j] * bf[j]);
    float r = 0.0f;
    if (kk > T) {
      r = in[j];
    } else if (kk == T) {
      if (atomicAdd(tie, 1u) < nties) r = in[j];
    }
    op[j] = r;
  }
  *(float4*)(out + e) = o;
}

// ---------------------------------------------------------------------------
// Host launcher
// ---------------------------------------------------------------------------
extern "C" void kernel_launch(void* const* d_in, const int* in_sizes, int n_in,
                              void* d_out, int out_size, void* d_ws, size_t ws_size,
                              hipStream_t stream) {
  (void)in_sizes; (void)n_in; (void)out_size; (void)ws_size;
  const float* x    = (const float*)d_in[0];
  const float* duty = (const float*)d_in[1];
  const float* bs   = (const float*)d_in[2];
  const int*   kp   = (const int*)d_in[3];
  uint32_t* ws  = (uint32_t*)d_ws;
  float*    out = (float*)d_out;

  kw_init<<<1, KW_THREADS, 0, stream>>>(duty, bs, kp, ws);

  // pass 0: key bits [31:21]  (2048 bins, no prefix yet)
  kw_hist<<<dim3(KW_SPLIT, KW_B), KW_THREADS, 0, stream>>>(x, ws, 21, 2047u, 0u);
  kw_select<<<KW_B, KW_THREADS, 0, stream>>>(ws, 21, 2048, 0);
  // pass 1: key bits [20:10]  (2048 bins, top-11-bit prefix)
  kw_hist<<<dim3(KW_SPLIT, KW_B), KW_THREADS, 0, stream>>>(x, ws, 10, 2047u,
                                                           0xFFE00000u);
  kw_select<<<KW_B, KW_THREADS, 0, stream>>>(ws, 10, 2048, 0);
  // pass 2: key bits [9:0]    (1024 bins, top-22-bit prefix) -> exact T
  kw_hist<<<dim3(KW_SPLIT, KW_B), KW_THREADS, 0, stream>>>(x, ws, 0, 1023u,
                                                           0xFFFFFC00u);
  kw_select<<<KW_B, KW_THREADS, 0, stream>>>(ws, 0, 1024, 1);

  kw_mask<<<dim3(KW_MASK_BLK, KW_B), KW_THREADS, 0, stream>>>(x, ws, out);
}